// LuongAttention_3667902070839
// MI455X (gfx1250) — compile-verified
//
#include <hip/hip_runtime.h>
#include <hip/hip_bf16.h>
#include <math.h>

// Problem dims (fixed by the reference)
#define B_   128
#define S_   4096
#define E_   512   // enc dim
#define D_   512   // dec dim
#define NEGV (-1e9f)

typedef __attribute__((ext_vector_type(2))) float v2f;
typedef __attribute__((ext_vector_type(8))) float v8f;

// ---------------------------------------------------------------------------
// Kernel 1: proj = dec_hidden [B, D] @ W^T  (W is [E, D] row-major)
// Real GEMM -> V_WMMA_F32_16X16X4_F32. One wave per 16x16 output tile.
// ---------------------------------------------------------------------------
__global__ __launch_bounds__(128) void
luong_proj_wmma(const float* __restrict__ dec,   // [B, D]
                const float* __restrict__ W,     // [E, D]
                float* __restrict__ proj)        // [B, E]
{
    const int NT = E_ / 16;                 // 32 tiles along E
    const int wave = threadIdx.x >> 5;      // 4 waves per block
    const int tile = blockIdx.x * 4 + wave; // 256 tiles total
    const int mt = tile / NT;               // 0..7
    const int nt = tile % NT;               // 0..31
    const int lane = threadIdx.x & 31;
    const int half = lane >> 4;             // 0 or 1
    const int r    = lane & 15;

    const int m0 = mt * 16;
    const int n0 = nt * 16;

    v8f acc = {};
    const float* arow = dec + (size_t)(m0 + r) * D_;
    const float* brow = W   + (size_t)(n0 + r) * D_;

#pragma unroll 4
    for (int k = 0; k < D_; k += 4) {
        v2f a, b;
        const float* ap = arow + k + 2 * half;
        const float* bp = brow + k + 2 * half;
        a.x = ap[0]; a.y = ap[1];
        b.x = bp[0]; b.y = bp[1];
        acc = __builtin_amdgcn_wmma_f32_16x16x4_f32(
            false, a, false, b, (short)0, acc, false, false);
    }

#pragma unroll
    for (int i = 0; i < 8; ++i) {
        const int row = i + 8 * half;
        proj[(size_t)(m0 + row) * E_ + n0 + r] = acc[i];
    }
}

// ---------------------------------------------------------------------------
// Kernel 2: fused scores + online softmax + context, single pass over enc.
// One 256-thread block (8 waves) per batch. Double-buffered 16-row enc tiles
// staged into LDS with GLOBAL_LOAD_ASYNC_TO_LDS_B128 (ASYNCcnt path):
// issue next tile -> wait asynccnt<=8 (in-order => current tile complete)
// -> barrier -> compute. End-of-iteration barrier fences buffer reuse.
// ---------------------------------------------------------------------------
#define TROWS 16
#define TLD   520            // padded row stride (floats): 16B aligned, %64 != 0
#define NTILE (S_ / TROWS)   // 256 tiles

__device__ __forceinline__ unsigned lds_addr32(const void* p) {
    // Shared-aperture flat address: addr[31:0] is the LDS byte offset.
    return (unsigned)(size_t)p;
}

__global__ __launch_bounds__(256) void
luong_stream_softmax_ctx(const float* __restrict__ enc,     // [B, S, E]
                         const unsigned char* __restrict__ mask, // [B, S]
                         const float* __restrict__ proj,    // [B, E]
                         float* __restrict__ attn_raw,      // [B, S] raw scores out
                         float* __restrict__ ctx,           // [B, E]
                         float* __restrict__ mstats)        // [B, 2] = {m, l}
{
    __shared__ float projS[E_];
    __shared__ float tile[2][TROWS * TLD];
    __shared__ float sc[TROWS];

    const int b    = blockIdx.x;
    const int tid  = threadIdx.x;
    const int wv   = tid >> 5;
    const int lane = tid & 31;
    const int row  = 2 * wv + (lane >> 4);  // 8 waves * 2 half-waves = 16 rows
    const int lsub = lane & 15;

    const size_t encBase = (size_t)b * S_ * E_;

    // Issue the 8 async b128 loads this thread owns for a 64-row x 128-float block.
    auto issue_tile = [&](int buf, int s0) {
#pragma unroll
        for (int j = 0; j < 8; ++j) {
            const int v  = tid + j * 256;   // 0..2047 float4 chunks
            const int rr = v >> 5;          // row (32 float4 per row)
            const int c4 = v & 31;          // float4 within row
            const float* g = enc + encBase + (size_t)(s0 + rr) * E_ + 4 * c4;
            const unsigned l = lds_addr32(&tile[buf][rr * TLD + 4 * c4]);
            asm volatile("global_load_async_to_lds_b128 %0, %1, off"
                         :: "v"(l), "v"(g) : "memory");
        }
    };

    // Prologue: kick off tile 0, stage proj[b] while it flies.
    issue_tile(0, 0);
    for (int i = tid; i < E_; i += 256) projS[i] = proj[(size_t)b * E_ + i];

    float m = -INFINITY;
    float l = 0.0f;
    float c0 = 0.0f, c1 = 0.0f;             // context elems e=tid, e=tid+256

    for (int it = 0; it < NTILE; ++it) {
        const int cur = it & 1;
        const int s0  = it * TROWS;

        if (it + 1 < NTILE) {
            issue_tile(cur ^ 1, s0 + TROWS);
            // In-order completion: <=8 outstanding => current tile landed.
            asm volatile("s_wait_asynccnt 0x8" ::: "memory");
        } else {
            asm volatile("s_wait_asynccnt 0x0" ::: "memory");
        }
        __syncthreads();   // publish current tile across all waves

        // Dot product: 16 lanes per row, 32 elements each
        float p = 0.0f;
        const float* trow = &tile[cur][row * TLD];
#pragma unroll 8
        for (int j = 0; j < 32; ++j) {
            const int e = lsub + 16 * j;
            p = fmaf(trow[e], projS[e], p);
        }
        p += __shfl_xor(p, 1, 32);
        p += __shfl_xor(p, 2, 32);
        p += __shfl_xor(p, 4, 32);
        p += __shfl_xor(p, 8, 32);

        if (lsub == 0) {
            const int sIdx = s0 + row;
            float scv = p;
            if (mask[(size_t)b * S_ + sIdx] == 0) scv = NEGV;
            sc[row] = scv;
            attn_raw[(size_t)b * S_ + sIdx] = scv;   // raw; K3 normalizes
        }
        __syncthreads();

        // Online softmax update (replicated per thread; all read same sc[])
        float tmax = sc[0];
#pragma unroll
        for (int i = 1; i < TROWS; ++i) tmax = fmaxf(tmax, sc[i]);
        const float newM = fmaxf(m, tmax);
        const float scale = __expf(m - newM);   // m=-inf first iter -> 0
        l *= scale; c0 *= scale; c1 *= scale;
#pragma unroll
        for (int i = 0; i < TROWS; ++i) {
            const float wgt = __expf(sc[i] - newM);
            l  += wgt;
            c0 = fmaf(wgt, tile[cur][i * TLD + tid],       c0);
            c1 = fmaf(wgt, tile[cur][i * TLD + tid + 256], c1);
        }
        m = newM;

        __syncthreads();   // fence: safe to overwrite tile[cur] next iteration
    }

    const float invl = 1.0f / l;
    ctx[(size_t)b * E_ + tid]       = c0 * invl;
    ctx[(size_t)b * E_ + tid + 256] = c1 * invl;
    if (tid == 0) { mstats[2 * b] = m; mstats[2 * b + 1] = l; }
}

// ---------------------------------------------------------------------------
// Kernel 3: normalize attention weights in place: exp(score - m) / l
// ---------------------------------------------------------------------------
__global__ __launch_bounds__(256) void
luong_attn_norm(float* __restrict__ attn,          // [B, S] raw -> normalized
                const float* __restrict__ mstats)  // [B, 2]
{
    const size_t idx = (size_t)blockIdx.x * 256 + threadIdx.x;
    const int b = (int)(idx >> 12);                // / S_ (4096)
    const float mv = mstats[2 * b];
    const float lv = mstats[2 * b + 1];
    const float s = attn[idx];
    attn[idx] = __expf(s - mv) * (1.0f / lv);
}

// ---------------------------------------------------------------------------
extern "C" void kernel_launch(void* const* d_in, const int* in_sizes, int n_in,
                              void* d_out, int out_size, void* d_ws, size_t ws_size,
                              hipStream_t stream) {
    (void)in_sizes; (void)n_in; (void)out_size; (void)ws_size;

    const float*         dec  = (const float*)d_in[0];          // [B, D]
    const float*         enc  = (const float*)d_in[1];          // [B, S, E]
    const unsigned char* mask = (const unsigned char*)d_in[2];  // [B, S] bool
    const float*         W    = (const float*)d_in[3];          // [E, D]

    float* attn = (float*)d_out;                   // first B*S floats
    float* ctx  = (float*)d_out + (size_t)B_ * S_; // next  B*E floats

    float* proj   = (float*)d_ws;                              // B*E floats (256 KB)
    float* mstats = (float*)d_ws + (size_t)B_ * E_;            // B*2 floats

    // K1: proj = dec @ W^T via WMMA fp32. 256 tiles, 4 waves/block.
    luong_proj_wmma<<<64, 128, 0, stream>>>(dec, W, proj);

    // K2: single streaming pass with async double-buffered LDS staging.
    luong_stream_softmax_ctx<<<B_, 256, 0, stream>>>(enc, mask, proj,
                                                     attn, ctx, mstats);

    // K3: normalize attn weights in place.
    luong_attn_norm<<<(B_ * S_) / 256, 256, 0, stream>>>(attn, mstats);
}